// VecKMSparse_15702400434756
// MI455X (gfx1250) — compile-verified
//
#include <hip/hip_runtime.h>
#include <math.h>

#define GH 480
#define GW 640
#define GD 128
#define KSZ 17
#define RAD 8

typedef __attribute__((ext_vector_type(2))) float v2f;
typedef __attribute__((ext_vector_type(8))) float v8f;

// ---------------- K0: init min slot ----------------
__global__ void vk_init_min(int* minslot) {
    *minslot = 0x7F7FFFFF;  // +FLT_MAX bits
}

// ---------------- K1: t_ref = min(t) ----------------
__global__ void vk_min(const float* __restrict__ t, int n, int* __restrict__ minslot) {
    float v = 3.4e38f;
    for (int i = blockIdx.x * blockDim.x + threadIdx.x; i < n; i += gridDim.x * blockDim.x)
        v = fminf(v, t[i]);
    #pragma unroll
    for (int off = 16; off > 0; off >>= 1)
        v = fminf(v, __shfl_xor(v, off, 32));
    if ((threadIdx.x & 31) == 0)
        atomicMin(minslot, __float_as_int(v));   // t >= 0 -> int order == float order
}

// ---------------- K2: bin events into complex grid ----------------
// block = 128 threads (one per channel d), grid-stride over events
__global__ void vk_bin(const float* __restrict__ t, const int* __restrict__ y,
                       const int* __restrict__ x, const float* __restrict__ Tv,
                       const float* __restrict__ tref_p,
                       float* __restrict__ grid, float* __restrict__ counts, int n) {
    const int d = threadIdx.x;
    const float tref = *tref_p;
    const float Td = Tv[d];
    for (int e = blockIdx.x; e < n; e += gridDim.x) {
        const float te = t[e];                  // e is uniform -> scalar loads
        const int cell = y[e] * GW + x[e];
        float s, c;
        __sincosf((te - tref) * Td, &s, &c);
        float* g = grid + ((size_t)cell * GD + d) * 2;
        atomicAdd(g,     c);
        atomicAdd(g + 1, s);
        if (d == 0) atomicAdd(counts + cell, 1.0f);
    }
}

// ---------------- K3: wx/wy tables via WMMA outer products ----------------
// wx[j,d] = exp(i*r_j*X_d), wy[i,d] = exp(i*r_i*Y_d), r_k = (k-8)/8.
// Rows 0..15 come from one 16x16 WMMA outer-product tile (r column x V row);
// row 16 has r=1.0 exactly, computed directly (avoids all masked stores).
__global__ void vk_weights(const float* __restrict__ Xv, const float* __restrict__ Yv,
                           float* __restrict__ wx, float* __restrict__ wy) {
    const int lane = threadIdx.x;          // one wave (32 lanes), EXEC all ones
    const int lo = lane & 15;
    const int hi = lane >> 4;
    // A (16x4): col K=0 = r values for rows 0..15; K=1..3 zero.
    // Lanes 16..31 carry the K=2 column of A -> must be zero.
    const float rv = ((float)lo - (float)RAD) * (1.0f / (float)RAD);
    v2f a;
    a.x = (lane < 16) ? rv : 0.0f;
    a.y = 0.0f;
    for (int tbl = 0; tbl < 2; ++tbl) {
        const float* V = tbl ? Yv : Xv;
        float* Wt      = tbl ? wy : wx;
        for (int nt = 0; nt < 8; ++nt) {       // 8 column tiles of 16 -> 128 channels
            v2f b;
            b.x = V[nt * 16 + lo];  // lanes>=16 hold B[K=2,:]; harmless (A[:,2]==0)
            b.y = 0.0f;
            v8f c = {};
            c = __builtin_amdgcn_wmma_f32_16x16x4_f32(
                    false, a, false, b, (short)0, c, false, false);
            // C layout: vgpr r, lane l -> M = r + 8*(l/16), N = l%16  (all rows valid)
            #pragma unroll
            for (int r = 0; r < 8; ++r) {
                const int row = r + 8 * hi;
                const int col = nt * 16 + lo;
                float s, co;
                __sincosf(c[r], &s, &co);
                Wt[(row * GD + col) * 2 + 0] = co;
                Wt[(row * GD + col) * 2 + 1] = s;
            }
        }
    }
    // row 16: r = 1.0 -> exp(i*V[d])
    for (int col = lane; col < GD; col += 32) {
        float s, c;
        __sincosf(Xv[col], &s, &c);
        wx[(16 * GD + col) * 2 + 0] = c;
        wx[(16 * GD + col) * 2 + 1] = s;
        __sincosf(Yv[col], &s, &c);
        wy[(16 * GD + col) * 2 + 0] = c;
        wy[(16 * GD + col) * 2 + 1] = s;
    }
}

// ---------------- K4: fused gather + separable window conv + rotation ----------------
// block = 128 threads (one per channel), one block per query
__global__ void vk_query(const float* __restrict__ grid, const float* __restrict__ counts,
                         const float* __restrict__ wx, const float* __restrict__ wy,
                         const int* __restrict__ qy, const int* __restrict__ qx,
                         const float* __restrict__ qt, const float* __restrict__ Tv,
                         const float* __restrict__ tref_p, float* __restrict__ out) {
    __shared__ float scnt[4];
    const int m = blockIdx.x;
    const int d = threadIdx.x;
    const int cy = qy[m], cx = qx[m];
    const bool interior = (cy >= RAD) & (cy < GH - RAD) & (cx >= RAD) & (cx < GW - RAD);

    // ---- window count sum (shared by all channels) ----
    float pcnt = 0.0f;
    if (interior) {
        const float* cb = counts + (size_t)(cy - RAD) * GW + (cx - RAD);
        for (int cIdx = d; cIdx < KSZ * KSZ; cIdx += 128) {
            const int i = cIdx / KSZ;
            const int j = cIdx - i * KSZ;
            pcnt += cb[(size_t)i * GW + j];
        }
    } else {
        for (int cIdx = d; cIdx < KSZ * KSZ; cIdx += 128) {
            const int i  = cIdx / KSZ;
            const int j  = cIdx - i * KSZ;
            const int gy = cy + i - RAD;
            const int gx = cx + j - RAD;
            if ((unsigned)gy < GH && (unsigned)gx < GW) pcnt += counts[gy * GW + gx];
        }
    }
    #pragma unroll
    for (int off = 16; off > 0; off >>= 1)
        pcnt += __shfl_xor(pcnt, off, 32);
    if ((d & 31) == 0) scnt[d >> 5] = pcnt;
    __syncthreads();
    const float cnt = scnt[0] + scnt[1] + scnt[2] + scnt[3];

    // ---- preload per-channel wx weights into registers (static indices) ----
    float wxr[KSZ], wxi[KSZ];
    #pragma unroll
    for (int j = 0; j < KSZ; ++j) {
        wxr[j] = wx[(j * GD + d) * 2 + 0];
        wxi[j] = wx[(j * GD + d) * 2 + 1];
    }

    // ---- separable-in-window complex accumulation ----
    float ar = 0.0f, ai = 0.0f;
    if (interior) {
        const size_t rowStride = (size_t)GW * GD * 2;
        const float* base = grid + (((size_t)(cy - RAD) * GW + (cx - RAD)) * GD + d) * 2;
        for (int i = 0; i < KSZ; ++i) {
            const float* rp = base + (size_t)i * rowStride;
            if (i + 1 < KSZ)
                __builtin_prefetch(rp + rowStride, 0, 0);   // global_prefetch_b8
            const float wyr = wy[(i * GD + d) * 2 + 0];
            const float wyi = wy[(i * GD + d) * 2 + 1];
            float tr = 0.0f, ti = 0.0f;
            #pragma unroll
            for (int j = 0; j < KSZ; ++j) {
                const float2 g = *(const float2*)(rp + (size_t)j * (GD * 2));
                tr = fmaf(g.x, wxr[j], fmaf(-g.y, wxi[j], tr));
                ti = fmaf(g.x, wxi[j], fmaf( g.y, wxr[j], ti));
            }
            ar = fmaf(tr, wyr, fmaf(-ti, wyi, ar));
            ai = fmaf(tr, wyi, fmaf( ti, wyr, ai));
        }
    } else {
        for (int i = 0; i < KSZ; ++i) {
            const int gy = cy + i - RAD;
            const float wyr = wy[(i * GD + d) * 2 + 0];
            const float wyi = wy[(i * GD + d) * 2 + 1];
            if ((unsigned)gy < GH) {
                const float* rp = grid + (((size_t)gy * GW) * GD + d) * 2;
                float tr = 0.0f, ti = 0.0f;
                #pragma unroll
                for (int j = 0; j < KSZ; ++j) {
                    const int gx = cx + j - RAD;
                    if ((unsigned)gx < GW) {
                        const float2 g = *(const float2*)(rp + (size_t)gx * (GD * 2));
                        tr = fmaf(g.x, wxr[j], fmaf(-g.y, wxi[j], tr));
                        ti = fmaf(g.x, wxi[j], fmaf( g.y, wxr[j], ti));
                    }
                }
                ar = fmaf(tr, wyr, fmaf(-ti, wyi, ar));
                ai = fmaf(tr, wyi, fmaf( ti, wyr, ai));
            }
        }
    }

    // ---- temporal re-centering + normalization ----
    const float tref = *tref_p;
    const float ang  = -(qt[m] - tref) * Tv[d];
    float s, c;
    __sincosf(ang, &s, &c);
    const float orr = ar * c - ai * s;
    const float oi  = ar * s + ai * c;
    const float scale = 11.313708498984761f / fmaxf(cnt, 1.0f);   // sqrt(128)/max(cnt,1)
    out[((size_t)m * GD + d) * 2 + 0] = orr * scale;
    out[((size_t)m * GD + d) * 2 + 1] = oi  * scale;
}

extern "C" void kernel_launch(void* const* d_in, const int* in_sizes, int n_in,
                              void* d_out, int out_size, void* d_ws, size_t ws_size,
                              hipStream_t stream) {
    const float* t   = (const float*)d_in[0];
    const int*   y   = (const int*)  d_in[1];
    const int*   x   = (const int*)  d_in[2];
    const int*   qy  = (const int*)  d_in[3];
    const int*   qx  = (const int*)  d_in[4];
    const float* qt  = (const float*)d_in[5];
    const float* Tv  = (const float*)d_in[6];
    const float* Xv  = (const float*)d_in[7];
    const float* Yv  = (const float*)d_in[8];
    const int n_events = in_sizes[0];
    const int M        = in_sizes[3];

    // workspace layout (floats)
    float*  ws     = (float*)d_ws;
    const size_t G = (size_t)GH * GW * GD * 2;       // complex grid
    float*  grid   = ws;
    float*  counts = ws + G;
    float*  wx     = counts + (size_t)GH * GW;
    float*  wy     = wx + (size_t)KSZ * GD * 2;
    int*    minsl  = (int*)(wy + (size_t)KSZ * GD * 2);

    // zero grid + counts every call (deterministic); capture-safe async memset
    hipMemsetAsync(grid, 0, (G + (size_t)GH * GW) * sizeof(float), stream);

    vk_init_min<<<1, 1, 0, stream>>>(minsl);
    vk_min     <<<512, 256, 0, stream>>>(t, n_events, minsl);
    vk_weights <<<1, 32, 0, stream>>>(Xv, Yv, wx, wy);
    vk_bin     <<<4096, 128, 0, stream>>>(t, y, x, Tv, (const float*)minsl,
                                          grid, counts, n_events);
    vk_query   <<<M, 128, 0, stream>>>(grid, counts, wx, wy, qy, qx, qt, Tv,
                                       (const float*)minsl, (float*)d_out);
}